// ListMLELoss_14035953123737
// MI455X (gfx1250) — compile-verified
//
#include <hip/hip_runtime.h>
#include <math.h>
#include <stdint.h>

#define N_ITEMS 2048
#define THREADS 1024   // 32 waves (wave32) per workgroup, one workgroup per row

// -------------------------------------------------------------------------
// Kernel 1: one block per row.
//  Phase 0: async-DMA scores row & mask row into LDS (ASYNCcnt path,
//           global_load_async_to_lds_b128) -- fully overlapped with sort.
//  Phase 1: build 64-bit sort keys from relevance, held in registers.
//           Lane l of wave w owns elements eA = w*64+l and eB = eA+32.
//  Phase 2: bitonic sort (descending):
//           j<=16 via __shfl_xor (in-wave), j==32 via local reg swap,
//           j>=64 (15 of 66 passes) via LDS exchange.
//  Phase 3: s_wait_asynccnt 0 + barrier, gather scores/mask by sorted index.
//  Phase 4: row max, e = exp(s - M), reverse inclusive Hillis-Steele scan.
//  Phase 5: loss contributions, block reduction, write per-row partial.
// -------------------------------------------------------------------------
__global__ __launch_bounds__(THREADS) void listmle_row_kernel(
    const float* __restrict__ scores,
    const float* __restrict__ relevance,
    const float* __restrict__ mask,
    float* __restrict__ row_loss)
{
    __shared__ unsigned long long keys[N_ITEMS];          // 16 KB (reused as eArr)
    __shared__ __align__(16) float ldsScores[N_ITEMS];    // 8 KB (async DMA target)
    __shared__ __align__(16) float ldsMask[N_ITEMS];      // 8 KB (async DMA target)
    __shared__ float red[32];

    const int tid = threadIdx.x;
    const int l   = tid & 31;          // lane (wave32)
    const int w   = tid >> 5;          // wave id 0..31
    const int eA  = w * 64 + l;        // element owned in reg A
    const int eB  = eA + 32;           // element owned in reg B
    const long long rowBase = (long long)blockIdx.x * N_ITEMS;

    // ---- Phase 0: issue async global->LDS copies (no VGPR round trip).
    // Waves 0..15 stage the scores row, waves 16..31 stage the mask row.
    if (tid < 512) {
        unsigned long long gbase =
            (unsigned long long)(uintptr_t)(scores + rowBase);
        unsigned int ldsaddr = (unsigned int)(uintptr_t)&ldsScores[tid * 4];
        unsigned int voff = (unsigned int)(tid * 16);
        asm volatile("global_load_async_to_lds_b128 %0, %1, %2"
                     :: "v"(ldsaddr), "v"(voff), "s"(gbase) : "memory");
    } else {
        int t = tid - 512;
        unsigned long long gbase =
            (unsigned long long)(uintptr_t)(mask + rowBase);
        unsigned int ldsaddr = (unsigned int)(uintptr_t)&ldsMask[t * 4];
        unsigned int voff = (unsigned int)(t * 16);
        asm volatile("global_load_async_to_lds_b128 %0, %1, %2"
                     :: "v"(ldsaddr), "v"(voff), "s"(gbase) : "memory");
    }

    // ---- Phase 1: build sort keys in registers (coalesced 128B/wave loads).
    unsigned long long rA, rB;
    {
        float ra = relevance[rowBase + eA];
        float rb = relevance[rowBase + eB];
        // Order-preserving mapping: float compare == unsigned compare.
        unsigned int u0 = __float_as_uint(ra);
        u0 ^= (u0 & 0x80000000u) ? 0xFFFFFFFFu : 0x80000000u;
        unsigned int u1 = __float_as_uint(rb);
        u1 ^= (u1 & 0x80000000u) ? 0xFFFFFFFFu : 0x80000000u;
        // Low 16 bits: (0xFFFF - idx) -> descending key sort is stable
        // (smaller original index first on ties), matching argsort(-rel).
        rA = ((unsigned long long)u0 << 32) | (unsigned int)(0xFFFF - eA);
        rB = ((unsigned long long)u1 << 32) | (unsigned int)(0xFFFF - eB);
    }

    // ---- Phase 2: bitonic sort, descending over element index.
    for (int k = 2; k <= N_ITEMS; k <<= 1) {
        // (a) cross-wave passes through LDS (j = k/2 .. 64), k >= 128 only
        for (int j = k >> 1; j >= 64; j >>= 1) {
            keys[eA] = rA;
            keys[eB] = rB;
            __syncthreads();
            unsigned long long pA = keys[eA ^ j];
            unsigned long long pB = keys[eB ^ j];
            // j>=64, k>=128: direction & lower/upper identical for eA and eB
            bool desc  = ((eA & k) == 0);
            bool lower = ((eA & j) == 0);
            bool takeMax = (lower == desc);
            rA = ((rA > pA) == takeMax) ? rA : pA;
            rB = ((rB > pB) == takeMax) ? rB : pB;
            __syncthreads();
        }
        // (b) j == 32: partner is the other register of the same lane
        if (k >= 64) {
            bool desc = ((eA & k) == 0);       // same for eA and eB (k>=64)
            unsigned long long mx = (rA > rB) ? rA : rB;
            unsigned long long mn = (rA > rB) ? rB : rA;
            rA = desc ? mx : mn;               // eA is the lower element
            rB = desc ? mn : mx;
        }
        // (c) in-wave passes via shuffle (j = min(k/2,16) .. 1)
        int jstart = ((k >> 1) < 16) ? (k >> 1) : 16;
        for (int j = jstart; j >= 1; j >>= 1) {
            bool descA = ((eA & k) == 0);
            bool descB = ((eB & k) == 0);
            bool lower = ((l & j) == 0);
            unsigned long long pA = __shfl_xor(rA, j, 32);
            unsigned long long pB = __shfl_xor(rB, j, 32);
            bool tmA = (lower == descA);
            bool tmB = (lower == descB);
            rA = ((rA > pA) == tmA) ? rA : pA;
            rB = ((rB > pB) == tmB) ? rB : pB;
        }
    }

    // ---- Phase 3: sorted indices straight from registers.
    const int idxA = 0xFFFF - (int)(rA & 0xFFFFu);
    const int idxB = 0xFFFF - (int)(rB & 0xFFFFu);

    // Wait for this wave's async LDS writes; the barrier makes completion
    // workgroup-wide and also fences the keys-region reuse (eArr) below.
    asm volatile("s_wait_asynccnt 0x0" ::: "memory");
    __syncthreads();

    float* eArr = (float*)keys;    // reuse 16 KB keys region (need 8 KB)

    float mA = ldsMask[idxA], mB = ldsMask[idxB];
    float sA = ldsScores[idxA], sB = ldsScores[idxB];
    if (mA == 0.0f) sA = -INFINITY;
    if (mB == 0.0f) sB = -INFINITY;

    // ---- Phase 4a: row max (shuffle + LDS tree; wave32).
    float mx = fmaxf(sA, sB);
    #pragma unroll
    for (int o = 16; o > 0; o >>= 1) mx = fmaxf(mx, __shfl_xor(mx, o, 32));
    if (l == 0) red[w] = mx;
    __syncthreads();
    if (tid < 32) {
        float v = red[tid];
        #pragma unroll
        for (int o = 16; o > 0; o >>= 1) v = fmaxf(v, __shfl_xor(v, o, 32));
        if (tid == 0) red[0] = v;
    }
    __syncthreads();
    float M = red[0];
    if (!isfinite(M)) M = 0.0f;    // fully-masked row guard
    __syncthreads();

    // ---- Phase 4b: e = exp(s - M); reverse inclusive scan (suffix sums).
    eArr[eA] = expf(sA - M);
    eArr[eB] = expf(sB - M);
    __syncthreads();
    for (int off = 1; off < N_ITEMS; off <<= 1) {
        float v0 = eArr[eA] + ((eA + off < N_ITEMS) ? eArr[eA + off] : 0.0f);
        float v1 = eArr[eB] + ((eB + off < N_ITEMS) ? eArr[eB + off] : 0.0f);
        __syncthreads();
        eArr[eA] = v0;
        eArr[eB] = v1;
        __syncthreads();
    }

    // ---- Phase 5: contributions over sorted positions [0, N-2].
    float c = 0.0f;
    if (mA != 0.0f) {                       // eA <= 2015 < N-1 always
        float lse = M + logf(eArr[eA]);
        c += (lse - sA) * mA;               // == -(s - rev_lse) * m
    }
    if (eB < N_ITEMS - 1 && mB != 0.0f) {
        float lse = M + logf(eArr[eB]);
        c += (lse - sB) * mB;
    }
    #pragma unroll
    for (int o = 16; o > 0; o >>= 1) c += __shfl_xor(c, o, 32);
    if (l == 0) red[w] = c;
    __syncthreads();
    if (tid < 32) {
        float v = red[tid];
        #pragma unroll
        for (int o = 16; o > 0; o >>= 1) v += __shfl_xor(v, o, 32);
        if (tid == 0) row_loss[blockIdx.x] = v;
    }
}

// -------------------------------------------------------------------------
// Kernel 2: deterministic mean over B per-row partials (single block, fixed
// reduction tree -> bit-identical across graph replays; no float atomics).
// -------------------------------------------------------------------------
__global__ __launch_bounds__(1024) void listmle_reduce_kernel(
    const float* __restrict__ part, float* __restrict__ out, int B)
{
    __shared__ float red[32];
    int tid = threadIdx.x;
    float s = 0.0f;
    for (int i = tid; i < B; i += 1024) s += part[i];
    #pragma unroll
    for (int o = 16; o > 0; o >>= 1) s += __shfl_xor(s, o, 32);
    if ((tid & 31) == 0) red[tid >> 5] = s;
    __syncthreads();
    if (tid < 32) {
        float v = red[tid];
        #pragma unroll
        for (int o = 16; o > 0; o >>= 1) v += __shfl_xor(v, o, 32);
        if (tid == 0) out[0] = v / (float)B;
    }
}

extern "C" void kernel_launch(void* const* d_in, const int* in_sizes, int n_in,
                              void* d_out, int out_size, void* d_ws, size_t ws_size,
                              hipStream_t stream)
{
    const float* scores    = (const float*)d_in[0];
    const float* relevance = (const float*)d_in[1];
    const float* mask      = (const float*)d_in[2];
    const int B = in_sizes[0] / N_ITEMS;           // 16384 rows of 2048

    float* rowLoss = (float*)d_ws;                 // B floats of scratch

    listmle_row_kernel<<<B, THREADS, 0, stream>>>(scores, relevance, mask, rowLoss);
    listmle_reduce_kernel<<<1, 1024, 0, stream>>>(rowLoss, (float*)d_out, B);
}